// WeatherAwareCricketGNN_17626545782988
// MI455X (gfx1250) — compile-verified
//
#include <hip/hip_runtime.h>
#include <hip/hip_bf16.h>

// ---------------------------------------------------------------------------
// WeatherAwareCricketGNN on MI455X (gfx1250).
//  * Dense matrix work (encoders, SAGE linear transforms) uses fp32 WMMA
//    (V_WMMA_F32_16X16X4_F32), K fully unrolled via templates so the per-lane
//    A strip is batch-loaded (no per-step s_wait_loadcnt 0x0 serialization).
//  * B weights staged in LDS in k-pair-interleaved layout -> each fragment
//    is one aligned ds_load_b64, no repack movs, conflict-free banks.
//  * Edge aggregation (the bandwidth-bound part): coalesced gathers + global
//    float atomics; edge indices scalarized via readfirstlane.
// ---------------------------------------------------------------------------

typedef __attribute__((ext_vector_type(2))) float v2f;
typedef __attribute__((ext_vector_type(8))) float v8f;

// ---------------------------------------------------------------------------
// Generic zero
// ---------------------------------------------------------------------------
__global__ void fzero_kernel(float* __restrict__ p, long long n) {
    long long i = (long long)blockIdx.x * blockDim.x + threadIdx.x;
    long long stride = (long long)gridDim.x * blockDim.x;
    for (; i < n; i += stride) p[i] = 0.0f;
}

// ---------------------------------------------------------------------------
// Edge scatter: agg[dst[e]][c] += x_src[src[e]][c];  cnt[dst[e]] += 1
// 64 consecutive threads (2 waves) share an edge: e is wave-uniform, so the
// src/dst index fetches are scalarized (s_load) via readfirstlane; the x_src
// gather and agg atomics stay coalesced 256B transactions.
// ---------------------------------------------------------------------------
__global__ void scatter_add_kernel(const float* __restrict__ xsrc,
                                   const int* __restrict__ src,
                                   const int* __restrict__ dst,
                                   float* __restrict__ agg,
                                   float* __restrict__ cnt,
                                   int nE) {
    long long t = (long long)blockIdx.x * blockDim.x + threadIdx.x;
    long long total = (long long)nE * 64;
    if (t >= total) return;
    int e = __builtin_amdgcn_readfirstlane((int)(t >> 6));  // wave-uniform
    int c = (int)(t & 63);
    int s = src[e];
    int d = dst[e];
    atomicAdd(&agg[(size_t)d * 64 + c], xsrc[(size_t)s * 64 + c]);
    if (c == 0) atomicAdd(&cnt[d], 1.0f);
}

// ---------------------------------------------------------------------------
// Fused WMMA GEMM (K0 = compile-time constant):
//   OUT[M x 64] (+)= (A[M x K0] * (cnt? 1/max(cnt,1) : 1)) @ B0[K0 x 64] + b0
//                    (+ X2[M x 64] @ B1[64 x 64]   when HAS_X2)
// One wave owns a 16(M) x 64(N) tile => 4 f32 16x16 accumulators.
// LDS B layout: pair-interleaved, idx = (k>>1)*130 + 2*n + (k&1) so a lane's
// {B[k][n], B[k+1][n]} fragment is a single 8B-aligned ds_load_b64; the
// 130-float pair-row stride keeps the upper half-wave on different banks.
// fp32 WMMA layout (16x16x4):
//   A frag: lane l holds row M=l&15, VGPR0/1 = K = kk + (l>=16?2:0) + {0,1}
//   B frag: lane l holds col N=l&15, VGPR0/1 = same K pair
//   C/D   : lane l, vgpr v -> M = v + (l>=16?8:0), N = l&15   (ISA 7.12.2)
// ---------------------------------------------------------------------------
#define GEMM_WAVES 8

template <int K0, bool HAS_X2>
__global__ void wmma_gemm_kernel(const float* __restrict__ A,
                                 const float* __restrict__ cnt,
                                 const float* __restrict__ X2,
                                 const float* __restrict__ B0,
                                 const float* __restrict__ b0,
                                 const float* __restrict__ B1,
                                 float* __restrict__ OUT,
                                 int M, int accumulate) {
    extern __shared__ float lds[];
    constexpr int NSTEP = K0 / 4;
    constexpr int BASE1 = (K0 / 2) * 130;  // B1 region (when HAS_X2)

    // Stage B0 (K0 x 64) and optional B1 (64 x 64) into LDS, pair-interleaved.
    for (int i = threadIdx.x; i < K0 * 64; i += blockDim.x) {
        int k = i >> 6, n = i & 63;
        lds[(k >> 1) * 130 + 2 * n + (k & 1)] = B0[i];
    }
    if (HAS_X2) {
        for (int i = threadIdx.x; i < 64 * 64; i += blockDim.x) {
            int k = i >> 6, n = i & 63;
            lds[BASE1 + (k >> 1) * 130 + 2 * n + (k & 1)] = B1[i];
        }
    }
    __syncthreads();

    int wave = threadIdx.x >> 5;
    int lane = threadIdx.x & 31;
    int lan15 = lane & 15;
    int ksel = (lane >> 4) * 2;               // 0 or 2
    int mbase = (blockIdx.x * GEMM_WAVES + wave) * 16;
    int mrow = mbase + lan15;
    bool mvalid = (mrow < M);
    int lrow = mvalid ? mrow : 0;             // clamped, always in-bounds

    v8f acc[4];
#pragma unroll
    for (int t = 0; t < 4; ++t) acc[t] = (v8f){0, 0, 0, 0, 0, 0, 0, 0};

    // ---- Pass 1: (A / max(cnt,1)) @ B0 -------------------------------------
    {
        const float* arow = A + (size_t)lrow * K0;
        float scale;
        if (cnt != nullptr)
            scale = mvalid ? (1.0f / fmaxf(cnt[mrow], 1.0f)) : 0.0f;
        else
            scale = mvalid ? 1.0f : 0.0f;

        v2f afrag[NSTEP];
#pragma unroll
        for (int s = 0; s < NSTEP; ++s)       // batched: all loads in flight
            afrag[s] = *(const v2f*)(arow + 4 * s + ksel);
#pragma unroll
        for (int s = 0; s < NSTEP; ++s) {
            afrag[s].x *= scale;
            afrag[s].y *= scale;
        }
#pragma unroll
        for (int s = 0; s < NSTEP; ++s) {
            int k0 = 4 * s + ksel;
#pragma unroll
            for (int nt = 0; nt < 4; ++nt) {
                v2f b = *(const v2f*)&lds[(k0 >> 1) * 130 + 2 * (nt * 16 + lan15)];
                acc[nt] = __builtin_amdgcn_wmma_f32_16x16x4_f32(
                    false, afrag[s], false, b, (short)0, acc[nt], false, false);
            }
        }
    }

    // ---- Pass 2: X2 @ B1 (K = 64) ------------------------------------------
    if (HAS_X2) {
        const float* xrow = X2 + (size_t)lrow * 64;
        float msk = mvalid ? 1.0f : 0.0f;
        v2f afrag[16];
#pragma unroll
        for (int s = 0; s < 16; ++s)
            afrag[s] = *(const v2f*)(xrow + 4 * s + ksel);
#pragma unroll
        for (int s = 0; s < 16; ++s) {
            afrag[s].x *= msk;
            afrag[s].y *= msk;
        }
#pragma unroll
        for (int s = 0; s < 16; ++s) {
            int k0 = 4 * s + ksel;
#pragma unroll
            for (int nt = 0; nt < 4; ++nt) {
                v2f b = *(const v2f*)&lds[BASE1 + (k0 >> 1) * 130 + 2 * (nt * 16 + lan15)];
                acc[nt] = __builtin_amdgcn_wmma_f32_16x16x4_f32(
                    false, afrag[s], false, b, (short)0, acc[nt], false, false);
            }
        }
    }

    // ---- Epilogue: + bias, optional accumulate into OUT --------------------
    int moff = (lane >> 4) * 8;
#pragma unroll
    for (int nt = 0; nt < 4; ++nt) {
        int col = nt * 16 + lan15;
        float bias = b0[col];
#pragma unroll
        for (int v = 0; v < 8; ++v) {
            int row = mbase + moff + v;
            if (row < M) {
                float val = acc[nt][v] + bias;
                float* p = OUT + (size_t)row * 64 + col;
                if (accumulate) *p += val; else *p = val;
            }
        }
    }
}

// ---------------------------------------------------------------------------
// x = relu(x * inv)  elementwise (mean over incoming edge-types + relu)
// ---------------------------------------------------------------------------
__global__ void relu_scale_kernel(float* __restrict__ p, long long n, float inv) {
    long long i = (long long)blockIdx.x * blockDim.x + threadIdx.x;
    long long stride = (long long)gridDim.x * blockDim.x;
    for (; i < n; i += stride) p[i] = fmaxf(p[i] * inv, 0.0f);
}

// ---------------------------------------------------------------------------
// Column sums of X[M x 64] into acc[64] (acc pre-zeroed).
// blockDim is a multiple of 64 so each thread stays on one column.
// ---------------------------------------------------------------------------
__global__ void colsum64_kernel(const float* __restrict__ X,
                                float* __restrict__ acc, long long M) {
    long long idx = (long long)blockIdx.x * blockDim.x + threadIdx.x;
    long long stride = (long long)gridDim.x * blockDim.x;
    int c = (int)(idx & 63);
    float s = 0.0f;
    for (long long t = idx; t < M * 64; t += stride) s += X[t];
    atomicAdd(&acc[c], s);
}

// ---------------------------------------------------------------------------
// Tail: all the small heads. Single block of 128 threads.
// Output layout: [0]=match_pred, [1..3]=weather_impact, [4..67]=player_emb,
// [68..131]=attended, [132..195]=venue_emb, [196..259]=team_emb
// ---------------------------------------------------------------------------
__global__ void tail_kernel(
    const float* __restrict__ pe_acc, float Pinv,
    const float* __restrict__ wf, const float* __restrict__ vf,
    const float* __restrict__ expfeat,
    const int* __restrict__ role_idx, const int* __restrict__ bat_idx,
    const int* __restrict__ bowl_idx,
    const float* __restrict__ Ww, const float* __restrict__ bw,
    const float* __restrict__ Wvn, const float* __restrict__ bvn,
    const float* __restrict__ role_tab, const float* __restrict__ bat_tab,
    const float* __restrict__ bowl_tab,
    const float* __restrict__ exp_W, const float* __restrict__ exp_b,
    const float* __restrict__ sq_W, const float* __restrict__ sq_b,
    const float* __restrict__ proj_W, const float* __restrict__ proj_b,
    const float* __restrict__ Wq, const float* __restrict__ bq,
    const float* __restrict__ Wk, const float* __restrict__ bk,
    const float* __restrict__ Wv, const float* __restrict__ bv,
    const float* __restrict__ Wo, const float* __restrict__ bo,
    const float* __restrict__ mp_W1, const float* __restrict__ mp_b1,
    const float* __restrict__ mp_W2, const float* __restrict__ mp_b2,
    const float* __restrict__ mp_W3, const float* __restrict__ mp_b3,
    const float* __restrict__ wi_W1, const float* __restrict__ wi_b1,
    const float* __restrict__ wi_W2, const float* __restrict__ wi_b2,
    float* __restrict__ out) {
    __shared__ float pe[64], we[64], ve[64], te[64];
    __shared__ float qv[64], kv[64], vv[64], att[64];
    __shared__ float h1[128], h2s[64], wih[32];
    __shared__ float scs[4], wsm[4];
    __shared__ float msq[2][32], sqe[2][32], pr[2][64];

    int tid = threadIdx.x;

    if (tid < 64) {
        pe[tid] = pe_acc[tid] * Pinv;
        float s = bw[tid];
        for (int i = 0; i < 16; ++i) s += wf[i] * Ww[i * 64 + tid];
        we[tid] = s;
        s = bvn[tid];
        for (int i = 0; i < 8; ++i) s += vf[i] * Wvn[i * 64 + tid];
        ve[tid] = s;
    }
    __syncthreads();

    // Squad features: mean over 11 players of [role|bat|bowl|exp@W+b] (32)
    if (tid < 64) {
        int s2 = tid >> 5, c = tid & 31;
        float acc = 0.0f;
        for (int p = 0; p < 11; ++p) {
            int pi = s2 * 11 + p;
            float fv;
            if (c < 8)       fv = role_tab[role_idx[pi] * 8 + c];
            else if (c < 16) fv = bat_tab[bat_idx[pi] * 8 + (c - 8)];
            else if (c < 24) fv = bowl_tab[bowl_idx[pi] * 8 + (c - 16)];
            else {
                int o = c - 24;
                fv = exp_b[o];
                for (int i = 0; i < 4; ++i) fv += expfeat[pi * 4 + i] * exp_W[i * 8 + o];
            }
            acc += fv;
        }
        msq[s2][c] = acc * (1.0f / 11.0f);
    }
    __syncthreads();
    if (tid < 64) {
        int s2 = tid >> 5, j = tid & 31;
        float s = sq_b[j];
        for (int c = 0; c < 32; ++c) s += msq[s2][c] * sq_W[c * 32 + j];
        sqe[s2][j] = s;
    }
    __syncthreads();
    if (tid < 128) {
        int s2 = tid >> 6, j = tid & 63;
        float s = proj_b[j];
        for (int c = 0; c < 32; ++c) s += sqe[s2][c] * proj_W[c * 64 + j];
        pr[s2][j] = s;
    }
    __syncthreads();
    if (tid < 64) te[tid] = 0.5f * (pr[0][tid] + pr[1][tid]);

    // Attention (softmax over a size-1 axis => weights == 1)
    if (tid < 64) {
        float q = bq[tid], k = bk[tid], v = bv[tid];
        for (int c = 0; c < 64; ++c) {
            q += pe[c] * Wq[c * 64 + tid];
            k += we[c] * Wk[c * 64 + tid];
            v += we[c] * Wv[c * 64 + tid];
        }
        qv[tid] = q; kv[tid] = k; vv[tid] = v;
    }
    __syncthreads();
    if (tid < 4) {
        float s = 0.0f;
        for (int d = 0; d < 16; ++d) s += qv[tid * 16 + d] * kv[tid * 16 + d];
        scs[tid] = s * 0.25f;     // /sqrt(16)
        wsm[tid] = 1.0f;          // softmax over singleton axis
    }
    __syncthreads();
    if (tid < 64) {
        float s = bo[tid];
        for (int c = 0; c < 64; ++c) s += (wsm[c >> 4] * vv[c]) * Wo[c * 64 + tid];
        att[tid] = s;
    }
    __syncthreads();

    // MLP head on combined = [pe | att | ve | te]
    if (tid < 128) {
        float s = mp_b1[tid];
        for (int c = 0; c < 64; ++c) s += pe[c] * mp_W1[c * 128 + tid];
        for (int c = 0; c < 64; ++c) s += att[c] * mp_W1[(64 + c) * 128 + tid];
        for (int c = 0; c < 64; ++c) s += ve[c] * mp_W1[(128 + c) * 128 + tid];
        for (int c = 0; c < 64; ++c) s += te[c] * mp_W1[(192 + c) * 128 + tid];
        h1[tid] = fmaxf(s, 0.0f);
    }
    __syncthreads();
    if (tid < 64) {
        float s = mp_b2[tid];
        for (int c = 0; c < 128; ++c) s += h1[c] * mp_W2[c * 64 + tid];
        h2s[tid] = fmaxf(s, 0.0f);
    }
    if (tid < 32) {
        float s = wi_b1[tid];
        for (int c = 0; c < 64; ++c) s += att[c] * wi_W1[c * 32 + tid];
        wih[tid] = fmaxf(s, 0.0f);
    }
    __syncthreads();
    if (tid == 0) {
        float s = mp_b3[0];
        for (int c = 0; c < 64; ++c) s += h2s[c] * mp_W3[c];
        out[0] = s;
    }
    if (tid < 3) {
        float s = wi_b2[tid];
        for (int j = 0; j < 32; ++j) s += wih[j] * wi_W2[j * 3 + tid];
        out[1 + tid] = s;
    }
    if (tid < 64) {
        out[4 + tid]   = pe[tid];
        out[68 + tid]  = att[tid];
        out[132 + tid] = ve[tid];
        out[196 + tid] = te[tid];
    }
    (void)scs;
}

// ---------------------------------------------------------------------------
// Host orchestration
// ---------------------------------------------------------------------------
extern "C" void kernel_launch(void* const* d_in, const int* in_sizes, int n_in,
                              void* d_out, int out_size, void* d_ws, size_t ws_size,
                              hipStream_t stream) {
    (void)n_in; (void)out_size; (void)ws_size;

    // Node types: 0=player 1=venue 2=team 3=match 4=weather
    const int NT[5] = {200000, 500, 200, 100000, 100000};
    const int FT[5] = {128, 16, 16, 32, 16};
    struct { int st, dt; } E[7] = {{0,0},{0,0},{0,1},{0,2},{3,1},{3,4},{2,3}};
    const float invE[5] = {0.5f, 0.5f, 1.0f, 1.0f, 1.0f};  // edges into each dst

    // ----- input pointers (setup_inputs order) -----
    const float* x_in[5];
    for (int t = 0; t < 5; ++t) x_in[t] = (const float*)d_in[t];
    const float* wf      = (const float*)d_in[5];
    const float* vf      = (const float*)d_in[6];
    const float* expfeat = (const float*)d_in[7];
    const int* srcp[7]; const int* dstp[7]; int nE[7];
    for (int i = 0; i < 7; ++i) {
        srcp[i] = (const int*)d_in[8 + 2 * i];
        dstp[i] = (const int*)d_in[9 + 2 * i];
        nE[i]   = in_sizes[8 + 2 * i];
    }
    const int* role_idx = (const int*)d_in[22];
    const int* bat_idx  = (const int*)d_in[23];
    const int* bowl_idx = (const int*)d_in[24];
    const float* Wenc[5]; const float* benc[5];
    for (int t = 0; t < 5; ++t) {
        Wenc[t] = (const float*)d_in[25 + 2 * t];
        benc[t] = (const float*)d_in[26 + 2 * t];
    }
    const float* sWl = (const float*)d_in[35];
    const float* sbl = (const float*)d_in[36];
    const float* sWr = (const float*)d_in[37];
    const float* Ww  = (const float*)d_in[38]; const float* bw  = (const float*)d_in[39];
    const float* Wvn = (const float*)d_in[40]; const float* bvn = (const float*)d_in[41];
    const float* role_tab = (const float*)d_in[42];
    const float* bat_tab  = (const float*)d_in[43];
    const float* bowl_tab = (const float*)d_in[44];
    const float* exp_W = (const float*)d_in[45]; const float* exp_b = (const float*)d_in[46];
    const float* sq_W  = (const float*)d_in[47]; const float* sq_b  = (const float*)d_in[48];
    const float* proj_W = (const float*)d_in[49]; const float* proj_b = (const float*)d_in[50];
    const float* Wq = (const float*)d_in[51]; const float* Wk = (const float*)d_in[52];
    const float* Wv = (const float*)d_in[53]; const float* Wo = (const float*)d_in[54];
    const float* bq = (const float*)d_in[55]; const float* bk = (const float*)d_in[56];
    const float* bv = (const float*)d_in[57]; const float* bo = (const float*)d_in[58];
    const float* mp_W1 = (const float*)d_in[59]; const float* mp_b1 = (const float*)d_in[60];
    const float* mp_W2 = (const float*)d_in[61]; const float* mp_b2 = (const float*)d_in[62];
    const float* mp_W3 = (const float*)d_in[63]; const float* mp_b3 = (const float*)d_in[64];
    const float* wi_W1 = (const float*)d_in[65]; const float* wi_b1 = (const float*)d_in[66];
    const float* wi_W2 = (const float*)d_in[67]; const float* wi_b2 = (const float*)d_in[68];

    // ----- workspace carve-up -----
    float* w = (float*)d_ws;
    size_t off = 0;
    auto alloc = [&](size_t n) { float* p = w + off; off += n; return p; };
    float* cur[5]; float* nxt[5];
    for (int t = 0; t < 5; ++t) cur[t] = alloc((size_t)NT[t] * 64);
    for (int t = 0; t < 5; ++t) nxt[t] = alloc((size_t)NT[t] * 64);
    float* agg   = alloc((size_t)NT[0] * 64);
    float* cnt   = alloc((size_t)NT[0]);
    float* peacc = alloc(64);

    auto zero = [&](float* p, long long n) {
        int blocks = (int)((n + 255) / 256);
        if (blocks > 65535) blocks = 65535;
        fzero_kernel<<<blocks, 256, 0, stream>>>(p, n);
    };

    // ----- encoders: cur[t] = x_in[t] @ Wenc[t] + benc[t]  (fp32 WMMA) -----
    for (int t = 0; t < 5; ++t) {
        int M = NT[t], K = FT[t];
        int grid = (M + 16 * GEMM_WAVES - 1) / (16 * GEMM_WAVES);
        size_t lds = (size_t)(K / 2) * 130 * sizeof(float);
        if (K == 128) {
            wmma_gemm_kernel<128, false><<<grid, 32 * GEMM_WAVES, lds, stream>>>(
                x_in[t], nullptr, nullptr, Wenc[t], benc[t], nullptr, cur[t], M, 0);
        } else if (K == 32) {
            wmma_gemm_kernel<32, false><<<grid, 32 * GEMM_WAVES, lds, stream>>>(
                x_in[t], nullptr, nullptr, Wenc[t], benc[t], nullptr, cur[t], M, 0);
        } else {  // K == 16
            wmma_gemm_kernel<16, false><<<grid, 32 * GEMM_WAVES, lds, stream>>>(
                x_in[t], nullptr, nullptr, Wenc[t], benc[t], nullptr, cur[t], M, 0);
        }
    }

    // ----- 3 SAGE layers -----
    for (int l = 0; l < 3; ++l) {
        for (int t = 0; t < 5; ++t) zero(nxt[t], (long long)NT[t] * 64);
        for (int i = 0; i < 7; ++i) {
            int st = E[i].st, dt = E[i].dt, nd = NT[dt];
            zero(agg, (long long)nd * 64);
            zero(cnt, (long long)nd);
            long long tot = (long long)nE[i] * 64;
            int sblocks = (int)((tot + 255) / 256);
            scatter_add_kernel<<<sblocks, 256, 0, stream>>>(
                cur[st], srcp[i], dstp[i], agg, cnt, nE[i]);
            int grid = (nd + 16 * GEMM_WAVES - 1) / (16 * GEMM_WAVES);
            size_t lds = (size_t)64 * 130 * sizeof(float);  // B0 (32 pairs) + B1 (32 pairs)
            wmma_gemm_kernel<64, true><<<grid, 32 * GEMM_WAVES, lds, stream>>>(
                agg, cnt, cur[dt],
                sWl + (size_t)(l * 7 + i) * 4096,
                sbl + (size_t)(l * 7 + i) * 64,
                sWr + (size_t)(l * 7 + i) * 4096,
                nxt[dt], nd, 1);
        }
        for (int t = 0; t < 5; ++t) {
            long long n = (long long)NT[t] * 64;
            int blocks = (int)((n + 255) / 256);
            if (blocks > 65535) blocks = 65535;
            relu_scale_kernel<<<blocks, 256, 0, stream>>>(nxt[t], n, invE[t]);
        }
        for (int t = 0; t < 5; ++t) { float* tmp = cur[t]; cur[t] = nxt[t]; nxt[t] = tmp; }
    }

    // ----- player_emb = mean over rows of final x_player -----
    zero(peacc, 64);
    colsum64_kernel<<<1024, 256, 0, stream>>>(cur[0], peacc, (long long)NT[0]);

    // ----- heads -----
    tail_kernel<<<1, 128, 0, stream>>>(
        peacc, 1.0f / (float)NT[0],
        wf, vf, expfeat, role_idx, bat_idx, bowl_idx,
        Ww, bw, Wvn, bvn, role_tab, bat_tab, bowl_tab,
        exp_W, exp_b, sq_W, sq_b, proj_W, proj_b,
        Wq, bq, Wk, bk, Wv, bv, Wo, bo,
        mp_W1, mp_b1, mp_W2, mp_b2, mp_W3, mp_b3,
        wi_W1, wi_b1, wi_W2, wi_b2,
        (float*)d_out);
}